// CTCLayer_35888746725988
// MI455X (gfx1250) — compile-verified
//
#include <hip/hip_runtime.h>
#include <hip/hip_bf16.h>
#include <stdint.h>

#define NEG_INF (-1e30f)
#define EPSV    (1e-7f)

typedef unsigned int v4u  __attribute__((ext_vector_type(4)));
typedef int          v8i_ __attribute__((ext_vector_type(8)));
typedef int          v4i_ __attribute__((ext_vector_type(4)));

// Issue one TDM 2-D tile load: tile_d1 rows of tensor_dim0 fp32 elements,
// contiguous rows (stride = tensor_dim0), global -> LDS. Tracked by TENSORcnt.
__device__ __forceinline__ void tdm_issue(void* lds_dst, const void* gsrc,
                                          int tensor_dim0, int tensor_dim1,
                                          int tile_d0, int tile_d1) {
    uint64_t ga = (uint64_t)(uintptr_t)gsrc;
    uint32_t lo = (uint32_t)(uintptr_t)lds_dst;   // low 32b of generic LDS addr = byte offset
    v4u g0;
    g0[0] = 1u;                                   // count=1, user descriptor
    g0[1] = lo;                                   // lds_addr (bytes)
    g0[2] = (uint32_t)ga;                         // global_addr[31:0]
    g0[3] = (uint32_t)((ga >> 32) & 0x1FFFFFFull) // global_addr[56:32]
            | (2u << 30);                         // type = 2 ("image")
    v8i_ g1;
    g1[0] = (2 << 16);                                            // data_size=2 -> 4B elems
    g1[1] = (int)(((unsigned)tensor_dim0 & 0xFFFFu) << 16);       // tensor_dim0[15:0]
    g1[2] = (int)((((unsigned)tensor_dim0 >> 16) & 0xFFFFu)
            | (((unsigned)tensor_dim1 & 0xFFFFu) << 16));         // dim0[31:16] | dim1[15:0]
    g1[3] = (int)((((unsigned)tensor_dim1 >> 16) & 0xFFFFu)
            | (((unsigned)tile_d0 & 0xFFFFu) << 16));             // dim1[31:16] | tile_dim0
    g1[4] = (int)((unsigned)tile_d1 & 0xFFFFu);                   // tile_dim1; tile_dim2=0
    g1[5] = tensor_dim0;                                          // tensor_dim0_stride[31:0]
    g1[6] = 0;                                                    // stride hi / dim1_stride lo
    g1[7] = 0;
    v4i_ z = {0, 0, 0, 0};                                        // groups 2/3: 2-D tensor
#if defined(__clang_major__) && (__clang_major__ >= 23)
    v8i_ z8 = {0, 0, 0, 0, 0, 0, 0, 0};
    __builtin_amdgcn_tensor_load_to_lds(g0, g1, z, z, z8, 0);
#else
    __builtin_amdgcn_tensor_load_to_lds(g0, g1, z, z, 0);
#endif
}

// One block per batch sample. 128 threads: thread s owns extended state s,
// thread 127 additionally owns state 128 (final blank). Alpha in LDS (double
// buffered, 2-entry NEG_INF halo). Probability rows stream via a 4-slot TDM
// ring: ROWS x C fp32 per chunk, up to NSLOTS-1 chunks in flight, drained
// with s_wait_tensorcnt so HBM/TDM latency is fully overlapped with the DP.
template <int T, int C, int L, int ROWS, int NSLOTS>
__global__ __launch_bounds__(128) void ctc_fwd_kernel(
    const int* __restrict__ y_true, const float* __restrict__ y_pred,
    const int* __restrict__ input_len, const int* __restrict__ label_len,
    float* __restrict__ out)
{
    constexpr int S  = 2 * L + 1;   // 129
    constexpr int NC = T / ROWS;    // chunks

    const int b   = blockIdx.x;
    const int tid = threadIdx.x;
    const int blank = C - 1;

    __shared__ float probs[NSLOTS][ROWS * C];  // 4 x 16 KB ring
    __shared__ float albuf[2][S + 3];          // state s at index s+2; [0..1] = -inf halo
    __shared__ int   labels[L];

    const float* base = y_pred + (size_t)b * T * C;

    // Prologue: wave 0 fills the ring (pipeline depth NSLOTS).
    if (tid < 32) {
        for (int c = 0; c < NSLOTS && c < NC; ++c)
            tdm_issue(&probs[c][0], base + (size_t)c * ROWS * C, C, T, C, ROWS);
    }

    if (tid < L) labels[tid] = y_true[b * L + tid];
    if (tid < 2) { albuf[0][tid] = NEG_INF; albuf[1][tid] = NEG_INF; }
    albuf[0][tid + 2] = (tid == 0) ? 0.0f : NEG_INF;   // alpha0
    if (tid == 127) albuf[0][S + 1] = NEG_INF;          // state 128
    __syncthreads();

    int  ext_s;
    bool allow2;
    {
        const int s = tid;
        const bool odd = (s & 1) != 0;
        ext_s  = odd ? labels[s >> 1] : blank;
        allow2 = odd && (s >= 2) && (labels[s >> 1] != labels[(s >> 1) - 1]);
    }
    const int ilen = input_len[b];
    const int eidx = 2 * label_len[b];   // end position (0..128)

    int cur = 0;
    for (int k = 0; k < NC; ++k) {
        // Chunk k must have landed; up to min(NSLOTS-1, NC-1-k) later chunks
        // may remain in flight (TENSORcnt completes in issue order).
        if (tid < 32) {
            const int rem = NC - 1 - k;     // chunks still to be consumed after k
            if      (rem >= 3) __builtin_amdgcn_s_wait_tensorcnt(3);
            else if (rem == 2) __builtin_amdgcn_s_wait_tensorcnt(2);
            else if (rem == 1) __builtin_amdgcn_s_wait_tensorcnt(1);
            else               __builtin_amdgcn_s_wait_tensorcnt(0);
        }
        __syncthreads();   // chunk k visible to all waves

        const float* prow = &probs[k % NSLOTS][0];
        for (int r = 0; r < ROWS; ++r) {
            const int t   = k * ROWS + r;
            const int nxt = cur ^ 1;

            // state tid: logsumexp(alpha[s], alpha[s-1], allow2?alpha[s-2]) + logp
            float a0 = albuf[cur][tid + 2];
            float a1 = albuf[cur][tid + 1];
            float a2 = allow2 ? albuf[cur][tid] : NEG_INF;
            float m  = fmaxf(a0, fmaxf(a1, a2));
            float sm = __expf(a0 - m) + __expf(a1 - m) + __expf(a2 - m);
            float lp = __logf(prow[r * C + ext_s] + EPSV);
            albuf[nxt][tid + 2] = m + __logf(sm) + lp;

            if (tid == 127) {   // state S-1 = 128 (blank, no skip)
                float b0 = albuf[cur][S + 1];
                float b1 = albuf[cur][S];
                float mm = fmaxf(b0, b1);
                float ss = __expf(b0 - mm) + __expf(b1 - mm);
                float lb = __logf(prow[r * C + blank] + EPSV);
                albuf[nxt][S + 1] = mm + __logf(ss) + lb;
            }
            __syncthreads();   // alpha_new visible

            if (t == ilen - 1 && tid == 0) {
                float e0 = albuf[nxt][eidx + 2];
                float e1 = albuf[nxt][eidx + 1];
                float me = fmaxf(e0, e1);
                out[b] = -(me + __logf(__expf(e0 - me) + __expf(e1 - me)));
            }
            cur = nxt;
        }
        __syncthreads();   // all waves done reading slot (k % NSLOTS)
        if (tid < 32 && (k + NSLOTS) < NC) {
            tdm_issue(&probs[k % NSLOTS][0],
                      base + (size_t)(k + NSLOTS) * ROWS * C, C, T, C, ROWS);
        }
    }
}

extern "C" void kernel_launch(void* const* d_in, const int* in_sizes, int n_in,
                              void* d_out, int out_size, void* d_ws, size_t ws_size,
                              hipStream_t stream) {
    const int*   y_true    = (const int*)d_in[0];
    const float* y_pred    = (const float*)d_in[1];
    const int*   input_len = (const int*)d_in[2];
    const int*   label_len = (const int*)d_in[3];
    float*       out       = (float*)d_out;

    const int B = in_sizes[2];   // input_len has B elements
    // Reference shapes: T=512, C=128, L=64; 32-row (16 KB) chunks, 4-slot ring.
    ctc_fwd_kernel<512, 128, 64, 32, 4><<<B, 128, 0, stream>>>(
        y_true, y_pred, input_len, label_len, out);
}